// Memory_2680059593218
// MI455X (gfx1250) — compile-verified
//
#include <hip/hip_runtime.h>
#include <hip/hip_bf16.h>
#include <stdint.h>

// ---------------------------------------------------------------------------
// DNC step, fused for MI455X (gfx1250, wave32).
// Dominant cost: one streaming read of link_matrix (268 MB) fused with both
// link contractions. Full-precision WMMA (f32_16x16x4_f32) for the interface
// GEMM; async-to-LDS loads (ASYNCcnt) for link-kernel staging.
// ---------------------------------------------------------------------------

#define B_   64
#define M_   1024
#define W_   64
#define R_   4
#define D_   512
#define IF_  471
#define IFP_ 480
#define DELTA_ 1e-6f

typedef __attribute__((ext_vector_type(2))) float v2f;
typedef __attribute__((ext_vector_type(8))) float v8f;

// workspace layout (float offsets)
#define OF_IFACE 0                         // 64*480
#define OF_NWW   (OF_IFACE + B_*IFP_)      // 64*1024
#define OF_CW    (OF_NWW   + B_*M_)        // 64*4*1024
#define OF_FW    (OF_CW    + B_*R_*M_)     // 64*4*1024
#define OF_BW    (OF_FW    + B_*R_*M_)     // 64*4*1024
#define OF_NM    (OF_BW    + B_*R_*M_)     // 64*1024*64

__device__ __forceinline__ float sigmoidf_(float x) { return 1.0f / (1.0f + expf(-x)); }
__device__ __forceinline__ float softplusf_(float x) {
    return (x > 20.0f) ? x : log1pf(expf(x));
}

// ---------------------------------------------------------------------------
// Kernel 1: iface = xi @ W + bW via V_WMMA_F32_16X16X4_F32 (full f32).
// One wave per 16x16 tile; K swept in steps of 4 (A: one float2 per lane).
// 32-bit A 16x4 layout: lane<16 -> K 0..1, lane>=16 -> K 2..3 (ISA 7.12.2).
// grid (30 n-tiles, 4 m-tiles), block = 32 threads (one wave32).
// ---------------------------------------------------------------------------
__global__ __launch_bounds__(32)
void k_iface_wmma(const float* __restrict__ xi, const float* __restrict__ W,
                  const float* __restrict__ bW, float* __restrict__ ifaceWs) {
    const int lane = threadIdx.x;          // 0..31
    const int half = lane >> 4;            // 0/1 -> K pair select
    const int sub  = lane & 15;            // 0..15
    const int rowA = blockIdx.y * 16 + sub;        // xi row (always < 64)
    const int colB = blockIdx.x * 16 + sub;        // iface col (may be >= 471)
    const float bmask = (colB < IF_) ? 1.0f : 0.0f;
    const int   colC  = (colB < IF_) ? colB : (IF_ - 1);   // clamped, branchless

    v8f c = {};
#pragma unroll 4
    for (int k0 = 0; k0 < D_; k0 += 4) {
        const int kk = k0 + half * 2;
        const v2f a = *(const v2f*)(xi + rowA * D_ + kk);          // aligned b64
        v2f b;
        b[0] = W[(kk + 0) * IF_ + colC] * bmask;
        b[1] = W[(kk + 1) * IF_ + colC] * bmask;
        c = __builtin_amdgcn_wmma_f32_16x16x4_f32(
                false, a, false, b, (short)0, c, false, false);
    }
#pragma unroll
    for (int v = 0; v < 8; ++v) {
        const int Mi = v + (half << 3);                 // tile row 0..15
        if (colB < IF_)
            ifaceWs[(blockIdx.y * 16 + Mi) * IFP_ + colB] = c[v] + bW[colB];
    }
}

// ---------------------------------------------------------------------------
// Kernel 2: per-batch DNC state update. 64 blocks x 1024 threads (m == tid).
// usage/psi, write-content softmax, allocation (stable bitonic sort +
// cumprod scan), new_write_w, new_memory rows fused with read-key logits
// and 4 content softmaxes.
// ---------------------------------------------------------------------------
__global__ __launch_bounds__(1024)
void k_state(const float* __restrict__ ifaceWs,
             const float* __restrict__ memory,
             const float* __restrict__ read_weights,
             const float* __restrict__ write_weights,
             const float* __restrict__ usage_vector,
             float* __restrict__ nwwWs, float* __restrict__ cwWs,
             float* __restrict__ nmWs) {
    __shared__ float sIf[IF_];
    __shared__ float sU[M_];
    __shared__ int   sIdx[M_];
    __shared__ float sP[M_];
    __shared__ float sRed[M_];
    __shared__ float sAlloc[M_];
    __shared__ float sWK[W_], sE[W_], sWV[W_], sRK[R_ * W_];

    const int t = threadIdx.x;       // == m row
    const int b = blockIdx.x;

    if (t < IF_) sIf[t] = ifaceWs[b * IFP_ + t];
    __syncthreads();

    if (t < W_) {
        sWK[t] = sIf[260 + t];
        sE[t]  = sigmoidf_(sIf[325 + t]);
        sWV[t] = sIf[389 + t];
    }
    if (t < R_ * W_) sRK[t] = sIf[t];   // read_keys = iface[:, 0:256]
    __syncthreads();

    // per-thread scalars (redundant, from LDS)
    const float ws_s = softplusf_(1.0f + fmaxf(sIf[324], 0.0f));
    const float ag   = sigmoidf_(sIf[457]);
    const float wg   = sigmoidf_(sIf[458]);
    float fg[R_], rs_s[R_];
#pragma unroll
    for (int r = 0; r < R_; ++r) {
        fg[r]   = sigmoidf_(sIf[453 + r]);
        rs_s[r] = softplusf_(1.0f + fmaxf(sIf[256 + r], 0.0f));
    }

    // --- usage update + retention ---
    float uu = usage_vector[b * M_ + t];
    const float wwold = write_weights[b * M_ + t];
    uu = uu + (1.0f - uu) * wwold;
    float psi = 1.0f;
#pragma unroll
    for (int r = 0; r < R_; ++r)
        psi *= (1.0f - fg[r] * read_weights[b * R_ * M_ + r * M_ + t]);
    uu *= psi;
    sU[t]   = DELTA_ + (1.0f - DELTA_) * uu;
    sIdx[t] = t;

    // --- write content weighting (cosine vs OLD memory, softmax over m) ---
    float nrm = 0.0f, dot = 0.0f, wk2 = 0.0f;
    const float* mrow = memory + ((size_t)b * M_ + t) * W_;
#pragma unroll 8
    for (int j = 0; j < W_; ++j) {
        const float mv = mrow[j];
        const float wk = sWK[j];
        nrm += mv * mv; dot += mv * wk; wk2 += wk * wk;
    }
    const float theta_w = dot / ((sqrtf(nrm) + DELTA_) * (sqrtf(wk2) + DELTA_));
    float logit = theta_w * ws_s;
    // block softmax
    sRed[t] = logit; __syncthreads();
    for (int s = 512; s > 0; s >>= 1) { if (t < s) sRed[t] = fmaxf(sRed[t], sRed[t + s]); __syncthreads(); }
    const float wmax = sRed[0]; __syncthreads();
    const float wexp = expf(logit - wmax);
    sRed[t] = wexp; __syncthreads();
    for (int s = 512; s > 0; s >>= 1) { if (t < s) sRed[t] += sRed[t + s]; __syncthreads(); }
    const float wcw = wexp / sRed[0]; __syncthreads();

    // --- allocation: stable bitonic sort (tie-break on index) ---
    for (unsigned k = 2; k <= M_; k <<= 1) {
        for (unsigned j = k >> 1; j > 0; j >>= 1) {
            const unsigned ixj = t ^ j;
            if (ixj > (unsigned)t) {
                const float a0 = sU[t],  a1 = sU[ixj];
                const int   i0 = sIdx[t], i1 = sIdx[ixj];
                const bool gt = (a0 > a1) || (a0 == a1 && i0 > i1);
                const bool asc = ((t & k) == 0);
                if (asc ? gt : !gt) {
                    sU[t] = a1; sU[ixj] = a0; sIdx[t] = i1; sIdx[ixj] = i0;
                }
            }
            __syncthreads();
        }
    }
    // inclusive cumprod scan of sorted u
    sP[t] = sU[t]; __syncthreads();
    for (int off = 1; off < M_; off <<= 1) {
        const float v = sP[t];
        const float p = (t >= off) ? sP[t - off] : 1.0f;
        __syncthreads();
        sP[t] = v * p;
        __syncthreads();
    }
    sAlloc[sIdx[t]] = (1.0f - sU[t]) * sP[t];
    __syncthreads();

    // --- new write weighting ---
    const float nww = wg * (ag * sAlloc[t] + (1.0f - ag) * wcw);
    nwwWs[b * M_ + t] = nww;

    // --- new memory row + read-key content logits ---
    float nrm2 = 0.0f;
    float dr[R_] = {0, 0, 0, 0}, rk2[R_] = {0, 0, 0, 0};
    float* nmrow = nmWs + ((size_t)b * M_ + t) * W_;
#pragma unroll 4
    for (int j = 0; j < W_; ++j) {
        const float nm = mrow[j] * (1.0f - nww * sE[j]) + nww * sWV[j];
        nmrow[j] = nm;
        nrm2 += nm * nm;
#pragma unroll
        for (int r = 0; r < R_; ++r) {
            const float rk = sRK[r * W_ + j];
            dr[r]  += nm * rk;
            rk2[r] += rk * rk;
        }
    }
    const float invn = 1.0f / (sqrtf(nrm2) + DELTA_);
#pragma unroll
    for (int r = 0; r < R_; ++r) {
        const float lr = dr[r] * invn / (sqrtf(rk2[r]) + DELTA_) * rs_s[r];
        sRed[t] = lr; __syncthreads();
        for (int s = 512; s > 0; s >>= 1) { if (t < s) sRed[t] = fmaxf(sRed[t], sRed[t + s]); __syncthreads(); }
        const float mx = sRed[0]; __syncthreads();
        const float e = expf(lr - mx);
        sRed[t] = e; __syncthreads();
        for (int s = 512; s > 0; s >>= 1) { if (t < s) sRed[t] += sRed[t + s]; __syncthreads(); }
        cwWs[b * R_ * M_ + r * M_ + t] = e / sRed[0];
        __syncthreads();
    }
}

// ---------------------------------------------------------------------------
// Kernel 3: fused link update + forward/backward contractions.
// new_link never materialized: lp = (1-ww_m-ww_n)*L[m,n] + ww_m*prec[n], diag 0.
// fw[r,m] = sum_n lp*rw[r,n] (register + wave reduce);
// bw[r,n] = sum_m rw[r,m]*lp (LDS float atomics, flushed via global atomics).
// ww/prec staged into LDS via async-to-LDS DMA (ASYNCcnt).
// grid (16 bands, 64 batches), 256 threads (8 waves x 8 rows each).
// ---------------------------------------------------------------------------
__global__ __launch_bounds__(256)
void k_link(const float* __restrict__ link,
            const float* __restrict__ prec,
            const float* __restrict__ read_weights,
            const float* __restrict__ nwwWs,
            float* __restrict__ fwWs, float* __restrict__ bwWs) {
    __shared__ float sRW4[M_ * R_];   // [n][r]
    __shared__ float sWW[M_];
    __shared__ float sPrec[M_];
    __shared__ float sBW[M_ * R_];    // [n][r] accumulator

    const int b    = blockIdx.y;
    const int base = blockIdx.x * 64;
    const int tid  = threadIdx.x;
    const int wave = tid >> 5, lane = tid & 31;

    for (int i = tid; i < M_; i += 256) {
        // ww / prec rows: async DMA directly into LDS (GV mode; low 32 bits of
        // a flat shared pointer are the LDS offset per the aperture mapping).
        {
            const unsigned dWW = (unsigned)(uintptr_t)&sWW[i];
            const unsigned dPr = (unsigned)(uintptr_t)&sPrec[i];
            const float* gWW = nwwWs + b * M_ + i;
            const float* gPr = prec  + b * M_ + i;
            asm volatile("global_load_async_to_lds_b32 %0, %1, off"
                         :: "v"(dWW), "v"(gWW) : "memory");
            asm volatile("global_load_async_to_lds_b32 %0, %1, off"
                         :: "v"(dPr), "v"(gPr) : "memory");
        }
#pragma unroll
        for (int r = 0; r < R_; ++r) {
            sRW4[i * R_ + r] = read_weights[b * R_ * M_ + r * M_ + i];
            sBW[i * R_ + r]  = 0.0f;
        }
    }
    asm volatile("s_wait_asynccnt 0x0" ::: "memory");
    __syncthreads();

    for (int rr = 0; rr < 8; ++rr) {
        const int m   = base + wave * 8 + rr;
        const float wwm = sWW[m];
        float rwm[R_];
#pragma unroll
        for (int r = 0; r < R_; ++r) rwm[r] = sRW4[m * R_ + r];

        float f[R_] = {0, 0, 0, 0};
        const float4* Lrow = (const float4*)(link + (size_t)b * M_ * M_ + (size_t)m * M_);
        for (int n4 = lane; n4 < M_ / 4; n4 += 32) {
            __builtin_prefetch(Lrow + n4 + 32, 0, 1);
            const float4 l4 = Lrow[n4];
            const float le[4] = {l4.x, l4.y, l4.z, l4.w};
#pragma unroll
            for (int e = 0; e < 4; ++e) {
                const int n = n4 * 4 + e;
                float lp = (1.0f - wwm - sWW[n]) * le[e] + wwm * sPrec[n];
                lp = (n == m) ? 0.0f : lp;
#pragma unroll
                for (int r = 0; r < R_; ++r) {
                    f[r] += lp * sRW4[n * R_ + r];
                    atomicAdd(&sBW[n * R_ + r], lp * rwm[r]);
                }
            }
        }
#pragma unroll
        for (int r = 0; r < R_; ++r) {
            for (int off = 16; off > 0; off >>= 1) f[r] += __shfl_down(f[r], off, 32);
            if (lane == 0) fwWs[b * R_ * M_ + r * M_ + m] = f[r];
        }
    }
    __syncthreads();
    for (int i = tid; i < M_ * R_; i += 256) {
        const int n = i >> 2, r = i & 3;
        atomicAdd(&bwWs[b * R_ * M_ + r * M_ + n], sBW[i]);
    }
}

// ---------------------------------------------------------------------------
// Kernel 4: read-mode softmax (over r, reference's exact slice), new_read_w,
// read_vectors = new_read_w @ new_memory. 64 blocks x 256 threads.
// ---------------------------------------------------------------------------
__global__ __launch_bounds__(256)
void k_readout(const float* __restrict__ ifaceWs,
               const float* __restrict__ fwWs, const float* __restrict__ bwWs,
               const float* __restrict__ cwWs, const float* __restrict__ nmWs,
               float* __restrict__ out) {
    __shared__ float sRM[R_ * 3];
    __shared__ float sNRW[R_ * M_];

    const int b = blockIdx.x, tid = threadIdx.x;

    if (tid == 0) {
        // read_modes = softmax over r of iface[:, 458:470].reshape(r,3)
        for (int c = 0; c < 3; ++c) {
            float v[R_], mx = -1e30f, s = 0.0f;
            for (int r = 0; r < R_; ++r) {
                v[r] = ifaceWs[b * IFP_ + 458 + r * 3 + c];
                mx = fmaxf(mx, v[r]);
            }
            for (int r = 0; r < R_; ++r) { v[r] = expf(v[r] - mx); s += v[r]; }
            for (int r = 0; r < R_; ++r) sRM[r * 3 + c] = v[r] / s;
        }
    }
    __syncthreads();

    for (int i = tid; i < R_ * M_; i += 256) {
        const int r = i >> 10, m = i & (M_ - 1);
        const int g = b * R_ * M_ + r * M_ + m;
        sNRW[i] = sRM[r * 3 + 0] * bwWs[g] + sRM[r * 3 + 1] * fwWs[g] +
                  sRM[r * 3 + 2] * cwWs[g];
    }
    __syncthreads();

    const int r = tid >> 6, j = tid & 63;
    const float* nm = nmWs + (size_t)b * M_ * W_ + j;
    float acc = 0.0f;
#pragma unroll 8
    for (int m = 0; m < M_; ++m)
        acc += sNRW[(r << 10) + m] * nm[m * W_];
    out[b * R_ * W_ + tid] = acc;
}

// ---------------------------------------------------------------------------
extern "C" void kernel_launch(void* const* d_in, const int* in_sizes, int n_in,
                              void* d_out, int out_size, void* d_ws, size_t ws_size,
                              hipStream_t stream) {
    const float* xi   = (const float*)d_in[0];
    const float* W    = (const float*)d_in[1];
    const float* bW   = (const float*)d_in[2];
    const float* mem  = (const float*)d_in[3];
    const float* link = (const float*)d_in[4];
    const float* prec = (const float*)d_in[5];
    const float* rw   = (const float*)d_in[6];
    const float* ww   = (const float*)d_in[7];
    const float* usg  = (const float*)d_in[8];
    float* out = (float*)d_out;
    float* ws  = (float*)d_ws;

    float* ifaceWs = ws + OF_IFACE;
    float* nwwWs   = ws + OF_NWW;
    float* cwWs    = ws + OF_CW;
    float* fwWs    = ws + OF_FW;
    float* bwWs    = ws + OF_BW;
    float* nmWs    = ws + OF_NM;

    // 1) interface GEMM (full-precision f32 WMMA)
    k_iface_wmma<<<dim3(IFP_ / 16, B_ / 16), 32, 0, stream>>>(xi, W, bW, ifaceWs);
    // 2) per-batch state update
    k_state<<<B_, M_, 0, stream>>>(ifaceWs, mem, rw, ww, usg, nwwWs, cwWs, nmWs);
    // 3) fused link pass (bw needs zero init)
    hipMemsetAsync(bwWs, 0, (size_t)B_ * R_ * M_ * sizeof(float), stream);
    k_link<<<dim3(M_ / 64, B_), 256, 0, stream>>>(link, prec, rw, nwwWs, fwWs, bwWs);
    // 4) readout
    k_readout<<<B_, 256, 0, stream>>>(ifaceWs, fwWs, bwWs, cwWs, nmWs, out);
}